// GAT_13280038879720
// MI455X (gfx1250) — compile-verified
//
#include <hip/hip_runtime.h>
#include <cstdint>

typedef __attribute__((ext_vector_type(2))) float v2f;
typedef __attribute__((ext_vector_type(8))) float v8f;

#define NEG_SLOPE 0.2f

// ---------------------------------------------------------------------------
// float atomic-max via integer punning (sign-aware; works for any IEEE f32)
// ---------------------------------------------------------------------------
__device__ __forceinline__ void atomicMaxFloat(float* addr, float val) {
  if (val >= 0.0f)
    atomicMax((int*)addr, __float_as_int(val));
  else
    atomicMin((unsigned int*)addr, (unsigned int)__float_as_int(val));
}

// ---------------------------------------------------------------------------
// f32 WMMA GEMM:  C[M,Nout] = A[M,K] x B[K,Nout]   (all row-major, f32)
// One wave computes a 16 x (NT*16) strip using V_WMMA_F32_16X16X4_F32.
// A fragment (16x4): lanes 0-15 -> rows, VGPR0/1 = K0/K1; lanes 16-31 = K2/K3
// B fragment (4x16): lanes 0-15 -> cols, VGPR0/1 = K0/K1; lanes 16-31 = K2/K3
// C (16x16, 8 VGPRs): vgpr v, lanes 0-15 -> M=v, lanes 16-31 -> M=v+8
// ---------------------------------------------------------------------------
template <int NT>
__global__ void gat_wmma_gemm_f32(const float* __restrict__ A,
                                  const float* __restrict__ B,
                                  float* __restrict__ C,
                                  int M, int K, int Nout) {
  const int wave = (int)((blockIdx.x * blockDim.x + threadIdx.x) >> 5);
  const int lane = (int)(threadIdx.x & 31);
  const int ntg  = Nout / (16 * NT);          // column groups per row-tile
  const int mt   = wave / ntg;                // 16-row tile index
  const int ng   = wave % ntg;                // column-group index
  if (mt * 16 >= M) return;                   // wave-uniform: EXEC stays full

  const int rrow   = mt * 16 + (lane & 15);
  const int row_a  = (rrow < M) ? rrow : (M - 1);
  const int koff   = (lane >> 4) * 2;         // 0 or 2
  const int colbase = ng * NT * 16 + (lane & 15);

  v8f acc[NT];
#pragma unroll
  for (int t = 0; t < NT; ++t) acc[t] = (v8f)0.0f;

  const float* __restrict__ arow = A + (size_t)row_a * K;
  for (int kk = 0; kk < K; kk += 4) {
    v2f a;
    a.x = arow[kk + koff + 0];
    a.y = arow[kk + koff + 1];
    const float* __restrict__ b0 = B + (size_t)(kk + koff) * Nout;
    const float* __restrict__ b1 = B + (size_t)(kk + koff + 1) * Nout;
#pragma unroll
    for (int t = 0; t < NT; ++t) {
      const int col = colbase + t * 16;
      v2f b;
      b.x = b0[col];
      b.y = b1[col];
      acc[t] = __builtin_amdgcn_wmma_f32_16x16x4_f32(
          /*neg_a=*/false, a, /*neg_b=*/false, b,
          /*c_mod=*/(short)0, acc[t], /*reuse_a=*/false, /*reuse_b=*/false);
    }
  }

#pragma unroll
  for (int t = 0; t < NT; ++t) {
    const int col = colbase + t * 16;
#pragma unroll
    for (int v = 0; v < 8; ++v) {
      const int row = mt * 16 + v + (lane >> 4) * 8;
      if (row < M) C[(size_t)row * Nout + col] = acc[t][v];
    }
  }
}

// ---------------------------------------------------------------------------
// alpha_src[n,h] = <h[n,h,:], a_src[h,:]>   (and same for a_dst)
// ---------------------------------------------------------------------------
__global__ void gat_alpha_kernel(const float* __restrict__ h,
                                 const float* __restrict__ a_src,
                                 const float* __restrict__ a_dst,
                                 float* __restrict__ asrc,
                                 float* __restrict__ adst,
                                 int N, int H, int F) {
  const int i = blockIdx.x * blockDim.x + threadIdx.x;
  if (i >= N * H) return;
  const int n = i / H, hd = i % H;
  const float* hp = h + (size_t)n * H * F + (size_t)hd * F;
  const float* as = a_src + (size_t)hd * F;
  const float* ad = a_dst + (size_t)hd * F;
  float s = 0.0f, d = 0.0f;
  for (int f = 0; f < F; ++f) {
    const float v = hp[f];
    s += v * as[f];
    d += v * ad[f];
  }
  asrc[i] = s;
  adst[i] = d;
}

// ---------------------------------------------------------------------------
// pass A: e = LeakyReLU(asrc[src]+adst[dst]); segment max into m[dst,h]
// ---------------------------------------------------------------------------
__global__ void gat_edge_max_kernel(const int* __restrict__ src,
                                    const int* __restrict__ dst,
                                    const float* __restrict__ asrc,
                                    const float* __restrict__ adst,
                                    float* __restrict__ m,
                                    int E, int H) {
  const int i = blockIdx.x * blockDim.x + threadIdx.x;
  if (i >= E * H) return;
  const int e = i / H, hd = i % H;
  const int s = src[e], d = dst[e];
  float v = asrc[s * H + hd] + adst[d * H + hd];
  v = (v > 0.0f) ? v : NEG_SLOPE * v;
  atomicMaxFloat(&m[d * H + hd], v);
}

// ---------------------------------------------------------------------------
// pass B (layer 1, H=8, F=32): one wave per edge, lane = feature.
// acc[dst,h,f] += exp(e-m)*h[src,h,f];  den[dst,h] += exp(e-m)  (lane h)
// ---------------------------------------------------------------------------
__global__ void gat_edge_acc8_kernel(const int* __restrict__ src,
                                     const int* __restrict__ dst,
                                     const float* __restrict__ asrc,
                                     const float* __restrict__ adst,
                                     const float* __restrict__ m,
                                     const float* __restrict__ h,
                                     float* __restrict__ acc,
                                     float* __restrict__ den,
                                     int E) {
  const int i = blockIdx.x * blockDim.x + threadIdx.x;
  if (i >= E * 32) return;
  const int e = i >> 5, f = i & 31;
  const int s = src[e], d = dst[e];
  const float* hs = h + (size_t)s * 256;
  float* ap = acc + (size_t)d * 256;
#pragma unroll
  for (int hd = 0; hd < 8; ++hd) {
    float v = asrc[s * 8 + hd] + adst[d * 8 + hd];
    v = (v > 0.0f) ? v : NEG_SLOPE * v;
    const float ex = __expf(v - m[d * 8 + hd]);
    atomicAdd(&ap[hd * 32 + f], ex * hs[hd * 32 + f]);
    if (f == hd) atomicAdd(&den[d * 8 + hd], ex);
  }
}

// ---------------------------------------------------------------------------
// pass B (layer 2, H=1, F=32)
// ---------------------------------------------------------------------------
__global__ void gat_edge_acc1_kernel(const int* __restrict__ src,
                                     const int* __restrict__ dst,
                                     const float* __restrict__ asrc,
                                     const float* __restrict__ adst,
                                     const float* __restrict__ m,
                                     const float* __restrict__ g,
                                     float* __restrict__ acc,
                                     float* __restrict__ den,
                                     int E) {
  const int i = blockIdx.x * blockDim.x + threadIdx.x;
  if (i >= E * 32) return;
  const int e = i >> 5, f = i & 31;
  const int s = src[e], d = dst[e];
  float v = asrc[s] + adst[d];
  v = (v > 0.0f) ? v : NEG_SLOPE * v;
  const float ex = __expf(v - m[d]);
  atomicAdd(&acc[(size_t)d * 32 + f], ex * g[(size_t)s * 32 + f]);
  if (f == 0) atomicAdd(&den[d], ex);
}

// ---------------------------------------------------------------------------
// out[n,c] = acc[n,c] / (den[n, c/F] + 1e-16) + bias[c]  (+ optional ELU)
// ---------------------------------------------------------------------------
__global__ void gat_norm_kernel(const float* __restrict__ acc,
                                const float* __restrict__ den,
                                const float* __restrict__ bias,
                                float* __restrict__ out,
                                int N, int H, int F, int apply_elu) {
  const int i = blockIdx.x * blockDim.x + threadIdx.x;
  const int C = H * F;
  if (i >= N * C) return;
  const int n = i / C, c = i % C;
  float v = acc[i] / (den[n * H + c / F] + 1e-16f) + bias[c];
  if (apply_elu) v = (v > 0.0f) ? v : (__expf(v) - 1.0f);
  out[i] = v;
}

// ---------------------------------------------------------------------------
extern "C" void kernel_launch(void* const* d_in, const int* in_sizes, int n_in,
                              void* d_out, int out_size, void* d_ws, size_t ws_size,
                              hipStream_t stream) {
  const float* x      = (const float*)d_in[0];
  const int*   ei     = (const int*)d_in[1];
  const float* W1     = (const float*)d_in[2];
  const float* a_src1 = (const float*)d_in[3];
  const float* a_dst1 = (const float*)d_in[4];
  const float* b1     = (const float*)d_in[5];
  const float* W2     = (const float*)d_in[6];
  const float* a_src2 = (const float*)d_in[7];
  const float* a_dst2 = (const float*)d_in[8];
  const float* b2     = (const float*)d_in[9];

  const int N = in_sizes[0] / 128;     // 50000
  const int E = in_sizes[1] / 2;       // 800000
  const int FIN = 128, C1 = 256, H1 = 8, F1 = 32, C2 = 32;
  const int* src = ei;
  const int* dst = ei + E;

  // ---- workspace carve-up (floats) ----
  float* ws = (float*)d_ws;
  size_t o = 0;
  float* h1    = ws + o; o += (size_t)N * C1;   // layer-1 features
  float* acc1  = ws + o; o += (size_t)N * C1;   // layer-1 accumulator -> h2
  float* asrc1 = ws + o; o += (size_t)N * H1;
  float* adst1 = ws + o; o += (size_t)N * H1;
  float* m1    = ws + o; o += (size_t)N * H1;
  float* den1  = ws + o; o += (size_t)N * H1;
  float* g2    = ws + o; o += (size_t)N * C2;   // layer-2 features
  float* acc2  = ws + o; o += (size_t)N * C2;
  float* asrc2 = ws + o; o += (size_t)N;
  float* adst2 = ws + o; o += (size_t)N;
  float* m2    = ws + o; o += (size_t)N;
  float* den2  = ws + o; o += (size_t)N;
  (void)ws_size; (void)n_in; (void)out_size;

  const int BLK = 256;
  const int NEG_INF_BITS = (int)0xFF800000u;

  // ---- per-call init (deterministic) ----
  hipMemsetAsync(acc1, 0, (size_t)N * C1 * sizeof(float), stream);
  hipMemsetAsync(den1, 0, (size_t)N * H1 * sizeof(float), stream);
  hipMemsetD32Async((hipDeviceptr_t)m1, NEG_INF_BITS, (size_t)N * H1, stream);
  hipMemsetAsync(acc2, 0, (size_t)N * C2 * sizeof(float), stream);
  hipMemsetAsync(den2, 0, (size_t)N * sizeof(float), stream);
  hipMemsetD32Async((hipDeviceptr_t)m2, NEG_INF_BITS, (size_t)N, stream);

  // ================= layer 1 =================
  {
    // GEMM1: h1[N,256] = x[N,128] @ W1[128,256]; NT=4 -> 64 cols/wave
    const int mtiles = (N + 15) / 16;
    const int waves  = mtiles * (C1 / 64);
    const int thr    = waves * 32;
    gat_wmma_gemm_f32<4><<<(thr + BLK - 1) / BLK, BLK, 0, stream>>>(
        x, W1, h1, N, FIN, C1);

    gat_alpha_kernel<<<(N * H1 + BLK - 1) / BLK, BLK, 0, stream>>>(
        h1, a_src1, a_dst1, asrc1, adst1, N, H1, F1);

    gat_edge_max_kernel<<<((size_t)E * H1 + BLK - 1) / BLK, BLK, 0, stream>>>(
        src, dst, asrc1, adst1, m1, E, H1);

    gat_edge_acc8_kernel<<<((size_t)E * 32 + BLK - 1) / BLK, BLK, 0, stream>>>(
        src, dst, asrc1, adst1, m1, h1, acc1, den1, E);

    // normalize + bias + ELU, in place: acc1 becomes h2
    gat_norm_kernel<<<((size_t)N * C1 + BLK - 1) / BLK, BLK, 0, stream>>>(
        acc1, den1, b1, acc1, N, H1, F1, /*elu=*/1);
  }

  // ================= layer 2 =================
  {
    // GEMM2: g2[N,32] = h2[N,256] @ W2[256,32]; NT=2 -> 32 cols/wave
    const int mtiles = (N + 15) / 16;
    const int waves  = mtiles * (C2 / 32);
    const int thr    = waves * 32;
    gat_wmma_gemm_f32<2><<<(thr + BLK - 1) / BLK, BLK, 0, stream>>>(
        acc1, W2, g2, N, C1, C2);

    gat_alpha_kernel<<<(N + BLK - 1) / BLK, BLK, 0, stream>>>(
        g2, a_src2, a_dst2, asrc2, adst2, N, 1, C2);

    gat_edge_max_kernel<<<((size_t)E + BLK - 1) / BLK, BLK, 0, stream>>>(
        src, dst, asrc2, adst2, m2, E, 1);

    gat_edge_acc1_kernel<<<((size_t)E * 32 + BLK - 1) / BLK, BLK, 0, stream>>>(
        src, dst, asrc2, adst2, m2, g2, acc2, den2, E);

    gat_norm_kernel<<<((size_t)N * C2 + BLK - 1) / BLK, BLK, 0, stream>>>(
        acc2, den2, b2, (float*)d_out, N, 1, C2, /*elu=*/0);
  }
}